// SelectiveStateSpaceModel_3582002725126
// MI455X (gfx1250) — compile-verified
//
#include <hip/hip_runtime.h>
#include <math.h>

// Problem constants (match reference)
#define NB   2
#define LL   2048
#define DD   2048
#define NS   16      // d_state
#define RK   64      // dt_rank
#define OO   96      // dt_rank + 2*d_state
#define NCH  16      // scan chunks
#define CLEN 128     // L / NCH

typedef __attribute__((ext_vector_type(2))) float v2f;
typedef __attribute__((ext_vector_type(8))) float v8f;

// ---------------------------------------------------------------------------
// GEMM1: dbc[BL, 96] = x[BL, 2048] @ W_in^T   (full fp32 via V_WMMA_F32_16X16X4_F32)
// One wave computes one 16x16 output tile.
// A-frag layout: lanes 0-15 hold M=lane, K=khalf+{0,1}; lanes 16-31 K=2+{0,1}.
// B-frag mirrors with N across lanes. C/D: vgpr r -> M = r + 8*(lane>=16), N = lane&15.
// ---------------------------------------------------------------------------
__global__ __launch_bounds__(32) void k_gemm_in(const float* __restrict__ x,
                                                const float* __restrict__ Win,
                                                float* __restrict__ dbc) {
  const int lane  = threadIdx.x;
  const int mt    = blockIdx.x;          // 0..255
  const int nt    = blockIdx.y;          // 0..5
  const int r16   = lane & 15;
  const int khalf = (lane >> 4) << 1;    // 0 or 2
  const float* ap = x   + (size_t)(mt * 16 + r16) * DD + khalf;
  const float* bp = Win + (size_t)(nt * 16 + r16) * DD + khalf;
  v8f acc = {};
  #pragma unroll 8
  for (int k = 0; k < DD; k += 4) {
    v2f a = *(const v2f*)(ap + k);
    v2f b = *(const v2f*)(bp + k);
    acc = __builtin_amdgcn_wmma_f32_16x16x4_f32(false, a, false, b, (short)0, acc,
                                                false, false);
  }
  const int mbase = mt * 16 + ((lane >> 4) << 3);
  const int n     = nt * 16 + r16;
  #pragma unroll
  for (int r = 0; r < 8; ++r)
    dbc[(size_t)(mbase + r) * OO + n] = acc[r];
}

// ---------------------------------------------------------------------------
// GEMM2: delta[BL, 2048] = softplus(dbc[:, :64] @ W_dt^T + b_dt)
// ---------------------------------------------------------------------------
__global__ __launch_bounds__(32) void k_gemm_dt(const float* __restrict__ dbc,
                                                const float* __restrict__ Wdt,
                                                const float* __restrict__ bdt,
                                                float* __restrict__ delta) {
  const int lane  = threadIdx.x;
  const int mt    = blockIdx.x;          // 0..255
  const int nt    = blockIdx.y;          // 0..127
  const int r16   = lane & 15;
  const int khalf = (lane >> 4) << 1;
  const float* ap = dbc + (size_t)(mt * 16 + r16) * OO + khalf;  // delta_lr cols 0..63
  const float* bp = Wdt + (size_t)(nt * 16 + r16) * RK + khalf;
  v8f acc = {};
  #pragma unroll
  for (int k = 0; k < RK; k += 4) {
    v2f a = *(const v2f*)(ap + k);
    v2f b = *(const v2f*)(bp + k);
    acc = __builtin_amdgcn_wmma_f32_16x16x4_f32(false, a, false, b, (short)0, acc,
                                                false, false);
  }
  const int mbase = mt * 16 + ((lane >> 4) << 3);
  const int n     = nt * 16 + r16;
  const float bias = bdt[n];
  #pragma unroll
  for (int r = 0; r < 8; ++r) {
    float v  = acc[r] + bias;
    float sp = (v > 20.f) ? v : log1pf(__expf(v));       // softplus
    delta[(size_t)(mbase + r) * DD + n] = sp;
  }
}

// ---------------------------------------------------------------------------
// Scan pass 1: per (b, d, chunk) compute chunk-local prod(exp(delta*A)) and
// chunk-local h (starting at 0). All 16 states live in registers.
// ---------------------------------------------------------------------------
__global__ __launch_bounds__(256) void k_scan_partial(
    const float* __restrict__ x, const float* __restrict__ delta,
    const float* __restrict__ dbc, const float* __restrict__ A_log,
    float* __restrict__ prodA, float* __restrict__ hend) {
  const int d = blockIdx.x * 256 + threadIdx.x;
  const int c = blockIdx.y;
  const int b = blockIdx.z;
  float A[NS], pa[NS], h[NS];
  const float4* arow = (const float4*)(A_log + (size_t)d * NS);
  #pragma unroll
  for (int q = 0; q < 4; ++q) {
    float4 al = arow[q];
    A[4*q+0] = -__expf(al.x); A[4*q+1] = -__expf(al.y);
    A[4*q+2] = -__expf(al.z); A[4*q+3] = -__expf(al.w);
  }
  #pragma unroll
  for (int n = 0; n < NS; ++n) { pa[n] = 1.f; h[n] = 0.f; }

  const int t0 = c * CLEN;
  for (int t = t0; t < t0 + CLEN; ++t) {
    const size_t row = (size_t)b * LL + t;
    const float dlt = delta[row * DD + d];
    const float xv  = x[row * DD + d];
    const float dx  = dlt * xv;
    const float4* brow = (const float4*)(dbc + row * OO + RK);   // B matrix row
    #pragma unroll
    for (int q = 0; q < 4; ++q) {
      float4 bm = brow[q];
      float e;
      e = __expf(dlt * A[4*q+0]); pa[4*q+0] *= e; h[4*q+0] = fmaf(e, h[4*q+0], dx * bm.x);
      e = __expf(dlt * A[4*q+1]); pa[4*q+1] *= e; h[4*q+1] = fmaf(e, h[4*q+1], dx * bm.y);
      e = __expf(dlt * A[4*q+2]); pa[4*q+2] *= e; h[4*q+2] = fmaf(e, h[4*q+2], dx * bm.z);
      e = __expf(dlt * A[4*q+3]); pa[4*q+3] *= e; h[4*q+3] = fmaf(e, h[4*q+3], dx * bm.w);
    }
  }
  const size_t base = (((size_t)b * DD + d) * NCH + c) * NS;
  float4* pp = (float4*)(prodA + base);
  float4* hp = (float4*)(hend  + base);
  #pragma unroll
  for (int q = 0; q < 4; ++q) {
    pp[q] = make_float4(pa[4*q], pa[4*q+1], pa[4*q+2], pa[4*q+3]);
    hp[q] = make_float4(h[4*q],  h[4*q+1],  h[4*q+2],  h[4*q+3]);
  }
}

// ---------------------------------------------------------------------------
// Scan pass 2: sequential carry across the 16 chunks (per (b,d) lane).
// hinit[b,d,c] = state entering chunk c.
// ---------------------------------------------------------------------------
__global__ __launch_bounds__(256) void k_scan_carry(
    const float* __restrict__ prodA, const float* __restrict__ hend,
    float* __restrict__ hinit) {
  const int g = blockIdx.x * 256 + threadIdx.x;   // b*DD + d
  float hc[NS];
  #pragma unroll
  for (int n = 0; n < NS; ++n) hc[n] = 0.f;
  for (int c = 0; c < NCH; ++c) {
    const size_t base = ((size_t)g * NCH + c) * NS;
    float4* out = (float4*)(hinit + base);
    const float4* pp = (const float4*)(prodA + base);
    const float4* hp = (const float4*)(hend  + base);
    #pragma unroll
    for (int q = 0; q < 4; ++q) {
      out[q] = make_float4(hc[4*q], hc[4*q+1], hc[4*q+2], hc[4*q+3]);
      float4 pa = pp[q]; float4 he = hp[q];
      hc[4*q+0] = fmaf(pa.x, hc[4*q+0], he.x);
      hc[4*q+1] = fmaf(pa.y, hc[4*q+1], he.y);
      hc[4*q+2] = fmaf(pa.z, hc[4*q+2], he.z);
      hc[4*q+3] = fmaf(pa.w, hc[4*q+3], he.w);
    }
  }
}

// ---------------------------------------------------------------------------
// Scan pass 3: re-run each chunk with the correct incoming state, emit
// y[b,t,d] = sum_n h*C + x*D.
// ---------------------------------------------------------------------------
__global__ __launch_bounds__(256) void k_scan_final(
    const float* __restrict__ x, const float* __restrict__ delta,
    const float* __restrict__ dbc, const float* __restrict__ A_log,
    const float* __restrict__ Dvec, const float* __restrict__ hinit,
    float* __restrict__ y) {
  const int d = blockIdx.x * 256 + threadIdx.x;
  const int c = blockIdx.y;
  const int b = blockIdx.z;
  float A[NS], h[NS];
  const float4* arow = (const float4*)(A_log + (size_t)d * NS);
  #pragma unroll
  for (int q = 0; q < 4; ++q) {
    float4 al = arow[q];
    A[4*q+0] = -__expf(al.x); A[4*q+1] = -__expf(al.y);
    A[4*q+2] = -__expf(al.z); A[4*q+3] = -__expf(al.w);
  }
  const float4* hp = (const float4*)(hinit + (((size_t)b * DD + d) * NCH + c) * NS);
  #pragma unroll
  for (int q = 0; q < 4; ++q) {
    float4 hv = hp[q];
    h[4*q+0] = hv.x; h[4*q+1] = hv.y; h[4*q+2] = hv.z; h[4*q+3] = hv.w;
  }
  const float Dd = Dvec[d];

  const int t0 = c * CLEN;
  for (int t = t0; t < t0 + CLEN; ++t) {
    const size_t row = (size_t)b * LL + t;
    const float dlt = delta[row * DD + d];
    const float xv  = x[row * DD + d];
    const float dx  = dlt * xv;
    const float4* brow = (const float4*)(dbc + row * OO + RK);        // B
    const float4* crow = (const float4*)(dbc + row * OO + RK + NS);   // C
    float acc = 0.f;
    #pragma unroll
    for (int q = 0; q < 4; ++q) {
      float4 bm = brow[q];
      float4 cm = crow[q];
      float e;
      e = __expf(dlt * A[4*q+0]); h[4*q+0] = fmaf(e, h[4*q+0], dx * bm.x); acc = fmaf(h[4*q+0], cm.x, acc);
      e = __expf(dlt * A[4*q+1]); h[4*q+1] = fmaf(e, h[4*q+1], dx * bm.y); acc = fmaf(h[4*q+1], cm.y, acc);
      e = __expf(dlt * A[4*q+2]); h[4*q+2] = fmaf(e, h[4*q+2], dx * bm.z); acc = fmaf(h[4*q+2], cm.z, acc);
      e = __expf(dlt * A[4*q+3]); h[4*q+3] = fmaf(e, h[4*q+3], dx * bm.w); acc = fmaf(h[4*q+3], cm.w, acc);
    }
    y[row * DD + d] = fmaf(xv, Dd, acc);
  }
}

// ---------------------------------------------------------------------------
extern "C" void kernel_launch(void* const* d_in, const int* in_sizes, int n_in,
                              void* d_out, int out_size, void* d_ws, size_t ws_size,
                              hipStream_t stream) {
  const float* x     = (const float*)d_in[0];
  const float* W_in  = (const float*)d_in[1];
  const float* W_dt  = (const float*)d_in[2];
  const float* b_dt  = (const float*)d_in[3];
  const float* A_log = (const float*)d_in[4];
  const float* Dv    = (const float*)d_in[5];
  float* y = (float*)d_out;

  // Workspace carve-up (floats): dbc 1.5MB, delta 33.5MB, scan state 3x4.2MB
  float* ws    = (float*)d_ws;
  float* dbc   = ws;
  float* delta = dbc   + (size_t)NB * LL * OO;
  float* prodA = delta + (size_t)NB * LL * DD;
  float* hend  = prodA + (size_t)NB * DD * NCH * NS;
  float* hinit = hend  + (size_t)NB * DD * NCH * NS;

  k_gemm_in   <<<dim3(NB * LL / 16, OO / 16), 32, 0, stream>>>(x, W_in, dbc);
  k_gemm_dt   <<<dim3(NB * LL / 16, DD / 16), 32, 0, stream>>>(dbc, W_dt, b_dt, delta);
  k_scan_partial<<<dim3(DD / 256, NCH, NB), 256, 0, stream>>>(x, delta, dbc, A_log,
                                                              prodA, hend);
  k_scan_carry<<<dim3(NB * DD / 256), 256, 0, stream>>>(prodA, hend, hinit);
  k_scan_final<<<dim3(DD / 256, NCH, NB), 256, 0, stream>>>(x, delta, dbc, A_log, Dv,
                                                            hinit, y);
}